// SGCN_56805237457082
// MI455X (gfx1250) — compile-verified
//
#include <hip/hip_runtime.h>
#include <hip/hip_bf16.h>

#define NN       102400      // total nodes
#define F_IN     400
#define HID      4
#define NGRAPH   256
#define NPG      400         // nodes per graph
#define F1       32          // fc1 width

typedef __attribute__((ext_vector_type(2))) float v2f;
typedef __attribute__((ext_vector_type(8))) float v8f;

// ---------------------------------------------------------------------------
// K1: h = X @ W  via V_WMMA_F32_16X16X4_F32.
// Each wave owns a 16-row tile of X; B = W^T zero-padded to 16 columns lives
// in LDS (16 x 400 floats, rows 4..15 zero) so the inner loop is:
//   global b64 (A float2) + ds b64 (B float2) + v_wmma_f32_16x16x4_f32.
// ---------------------------------------------------------------------------
__global__ void xw_wmma_kernel(const float* __restrict__ X,
                               const float* __restrict__ W,
                               float* __restrict__ h) {
    __shared__ float Wt[16 * F_IN];          // Wt[n*400 + k] = (n<4)? W[k][n] : 0
    for (int i = threadIdx.x; i < 16 * F_IN; i += blockDim.x) {
        const int n = i / F_IN;
        const int k = i - n * F_IN;
        Wt[i] = (n < HID) ? W[k * HID + n] : 0.0f;
    }
    __syncthreads();

    const int lane  = threadIdx.x & 31;
    const int wave  = threadIdx.x >> 5;
    const int tile  = blockIdx.x * (blockDim.x >> 5) + wave;   // 16-row tile id
    const int row0  = tile * 16;
    const int m     = lane & 15;                // M (for A) / N (for B)
    const int khalf = (lane >> 4) << 1;         // 0 or 2

    const float* arow = X + (size_t)(row0 + m) * F_IN + khalf;
    const float* brow = Wt + m * F_IN + khalf;

    v8f c = {};
    #pragma unroll 4
    for (int k = 0; k < F_IN; k += 4) {
        v2f a = *reinterpret_cast<const v2f*>(arow + k);   // A: X[row0+m][k+khalf .. +1]
        v2f b = *reinterpret_cast<const v2f*>(brow + k);   // B: W^T[m][k+khalf .. +1]
        // (neg_a, A, neg_b, B, c_mod, C, reuse_a, reuse_b)
        c = __builtin_amdgcn_wmma_f32_16x16x4_f32(false, a, false, b,
                                                  (short)0, c, false, false);
    }

    // D layout: VGPR v -> row (v + 8*(lane>=16)), col (lane&15). Keep cols 0..3.
    if (m < HID) {
        const int rbase = row0 + ((lane >> 4) << 3);
        #pragma unroll
        for (int v = 0; v < 8; ++v)
            h[(size_t)(rbase + v) * HID + m] = c[v];
    }
}

// ---------------------------------------------------------------------------
// K2a: deg[i] = 1.0 (self-loop weight);  K2b: deg[dst[e]] += w[e]
// ---------------------------------------------------------------------------
__global__ void deg_init_kernel(float* __restrict__ deg, int n) {
    int i = blockIdx.x * blockDim.x + threadIdx.x;
    if (i < n) deg[i] = 1.0f;
}

__global__ void deg_scatter_kernel(const long long* __restrict__ dst,
                                   const float* __restrict__ ew,
                                   float* __restrict__ deg, int E) {
    int e = blockIdx.x * blockDim.x + threadIdx.x;
    if (e < E) atomicAdd(&deg[dst[e]], ew[e]);
}

// ---------------------------------------------------------------------------
// K3: dinv = rsqrt(deg); fold self-loop term: agg[i] = dinv[i]^2 * h[i]
// ---------------------------------------------------------------------------
__global__ void dinv_self_kernel(const float* __restrict__ deg,
                                 const float* __restrict__ h,
                                 float* __restrict__ dinv,
                                 float* __restrict__ agg, int n) {
    int i = blockIdx.x * blockDim.x + threadIdx.x;
    if (i < n) {
        float d  = deg[i];
        float di = (d > 0.0f) ? rsqrtf(d) : 0.0f;
        dinv[i]  = di;
        float s  = di * di;
        float4 hv = *reinterpret_cast<const float4*>(h + (size_t)i * HID);
        float4 o  = make_float4(s * hv.x, s * hv.y, s * hv.z, s * hv.w);
        *reinterpret_cast<float4*>(agg + (size_t)i * HID) = o;
    }
}

// ---------------------------------------------------------------------------
// K4: agg[dst] += dinv[src]*w*dinv[dst] * h[src]   (h/agg are L2-resident)
// ---------------------------------------------------------------------------
__global__ void edge_scatter_kernel(const long long* __restrict__ src,
                                    const long long* __restrict__ dst,
                                    const float* __restrict__ ew,
                                    const float* __restrict__ dinv,
                                    const float* __restrict__ h,
                                    float* __restrict__ agg, int E) {
    int e = blockIdx.x * blockDim.x + threadIdx.x;
    if (e < E) {
        long long s = src[e];
        long long t = dst[e];
        float nrm = dinv[s] * ew[e] * dinv[t];
        float4 hv = *reinterpret_cast<const float4*>(h + (size_t)s * HID);
        float* o  = agg + (size_t)t * HID;
        atomicAdd(o + 0, nrm * hv.x);
        atomicAdd(o + 1, nrm * hv.y);
        atomicAdd(o + 2, nrm * hv.z);
        atomicAdd(o + 3, nrm * hv.w);
    }
}

// ---------------------------------------------------------------------------
// K5: fc1: out[g][n] = relu( sum_k (agg_flat[g*1600+k] + b[k%4]) * w1[k][n] + b1[n] )
// One graph per block, one output column per lane (broadcast activation reads,
// coalesced weight reads; w1 = 204 KB -> L2 resident).
// ---------------------------------------------------------------------------
__global__ void fc1_kernel(const float* __restrict__ agg,
                           const float* __restrict__ bvec,
                           const float* __restrict__ w1,
                           const float* __restrict__ b1,
                           float* __restrict__ out) {
    const int g = blockIdx.x;       // 0..255
    const int n = threadIdx.x;      // 0..31
    const float bb0 = bvec[0], bb1 = bvec[1], bb2 = bvec[2], bb3 = bvec[3];
    const float* arow = agg + (size_t)g * (NPG * HID);
    float acc = b1[n];
    #pragma unroll 4
    for (int k = 0; k < NPG * HID; k += 4) {
        float4 av = *reinterpret_cast<const float4*>(arow + k);
        acc += (av.x + bb0) * w1[(k + 0) * F1 + n];
        acc += (av.y + bb1) * w1[(k + 1) * F1 + n];
        acc += (av.z + bb2) * w1[(k + 2) * F1 + n];
        acc += (av.w + bb3) * w1[(k + 3) * F1 + n];
    }
    out[g * F1 + n] = fmaxf(acc, 0.0f);
}

// ---------------------------------------------------------------------------
// K6: fc2 logits [256,2] + reg scalar (= 0) at d_out[512]
// ---------------------------------------------------------------------------
__global__ void fc2_kernel(const float* __restrict__ h1,
                           const float* __restrict__ w2,
                           const float* __restrict__ b2,
                           float* __restrict__ out) {
    int t = blockIdx.x * blockDim.x + threadIdx.x;
    if (t < NGRAPH * 2) {
        int g = t >> 1, c = t & 1;
        float acc = b2[c];
        const float* hrow = h1 + g * F1;
        #pragma unroll
        for (int j = 0; j < F1; ++j) acc += hrow[j] * w2[j * 2 + c];
        out[t] = acc;
    } else if (t == NGRAPH * 2) {
        out[NGRAPH * 2] = 0.0f;   // reg
    }
}

// ---------------------------------------------------------------------------
extern "C" void kernel_launch(void* const* d_in, const int* in_sizes, int n_in,
                              void* d_out, int out_size, void* d_ws, size_t ws_size,
                              hipStream_t stream) {
    const float*     x  = (const float*)d_in[0];     // [N, 400]
    const long long* ei = (const long long*)d_in[1]; // [2, E] int64
    const float*     ea = (const float*)d_in[2];     // [E]
    const float*     W  = (const float*)d_in[5];     // [400, 4]
    const float*     bv = (const float*)d_in[6];     // [4]
    const float*     w1 = (const float*)d_in[7];     // [1600, 32]
    const float*     b1 = (const float*)d_in[8];     // [32]
    const float*     w2 = (const float*)d_in[9];     // [32, 2]
    const float*     b2 = (const float*)d_in[10];    // [2]
    float* out = (float*)d_out;

    const int E = in_sizes[2];
    const long long* esrc = ei;
    const long long* edst = ei + E;

    // workspace carve-up (~4.2 MB)
    float* h    = (float*)d_ws;           // N*4
    float* deg  = h    + (size_t)NN * HID;  // N
    float* dinv = deg  + NN;                // N
    float* agg  = dinv + NN;                // N*4
    float* fc1o = agg  + (size_t)NN * HID;  // 256*32

    // K1: h = X @ W (WMMA f32 16x16x4). 8 waves/block * 16 rows = 128 rows/block.
    xw_wmma_kernel<<<NN / 128, 256, 0, stream>>>(x, W, h);

    // K2: degrees (self loop = 1, then edge weights)
    deg_init_kernel<<<(NN + 255) / 256, 256, 0, stream>>>(deg, NN);
    deg_scatter_kernel<<<(E + 255) / 256, 256, 0, stream>>>(edst, ea, deg, E);

    // K3: dinv + self-loop contribution into agg
    dinv_self_kernel<<<(NN + 255) / 256, 256, 0, stream>>>(deg, h, dinv, agg, NN);

    // K4: normalized edge scatter-add
    edge_scatter_kernel<<<(E + 255) / 256, 256, 0, stream>>>(esrc, edst, ea, dinv, h, agg, E);

    // K5: bias + fc1 + relu
    fc1_kernel<<<NGRAPH, F1, 0, stream>>>(agg, bv, w1, b1, fc1o);

    // K6: fc2 + reg
    fc2_kernel<<<3, 256, 0, stream>>>(fc1o, w2, b2, out);
}